// GATConv_35820027248978
// MI455X (gfx1250) — compile-verified
//
#include <hip/hip_runtime.h>

#define NODES 8192
#define FDIM  128
#define NEDGE 262144
#define ALPHA 0.2f

typedef __attribute__((ext_vector_type(2))) float v2f;
typedef __attribute__((ext_vector_type(8))) float v8f;

// ---------- ordered-float key for atomicMax on unsigned ----------
__device__ __forceinline__ unsigned fkey(float f) {
    unsigned u = __float_as_uint(f);
    return (u & 0x80000000u) ? ~u : (u | 0x80000000u);
}
__device__ __forceinline__ float fdec(unsigned k) {
    unsigned u = (k & 0x80000000u) ? (k & 0x7FFFFFFFu) : ~k;
    return __uint_as_float(u);
}

// ---------- 0) zero accumulators / sentinels ----------
__global__ void gat_zero(float* out, unsigned* rowmax, float* rowsum, float* colsum) {
    int i = blockIdx.x * blockDim.x + threadIdx.x;
    if (i < NODES * FDIM) out[i] = 0.0f;
    if (i < NODES) { rowmax[i] = 0u; rowsum[i] = 0.0f; }
    if (i < FDIM)  colsum[i] = 0.0f;
}

// ---------- 1) h = x @ W via V_WMMA_F32_16X16X4_F32 ----------
// Block = 128 threads (4 waves). Each wave computes a 16-row strip across all
// 128 output columns (8 tiles of 16x16, v8f accumulators).
// W is staged in LDS in *fragment order*: K-pair interleaved, so each B
// fragment is a single aligned ds_load_b64 into an even VGPR pair (no repack
// movs feeding the WMMAs).
__global__ void __launch_bounds__(128) gat_gemm(const float* __restrict__ x,
                                                const float* __restrict__ W,
                                                float* __restrict__ h) {
    // Wp[kp*128 + n] = { W[2kp][n], W[2kp+1][n] }   (64 KB of 320 KB WGP LDS)
    __shared__ __align__(16) v2f Wp[(FDIM / 2) * FDIM];
    const int tid = threadIdx.x;

    for (int i = tid; i < (FDIM / 2) * FDIM; i += 128) {
        const int kp = i >> 7;        // K-pair index
        const int n  = i & (FDIM - 1);
        v2f t;
        t.x = W[(2 * kp)     * FDIM + n];   // coalesced b32 across threads
        t.y = W[(2 * kp + 1) * FDIM + n];
        Wp[i] = t;
    }
    __syncthreads();

    const int waveId = tid >> 5;
    const int lane   = tid & 31;
    const int m  = lane & 15;   // row within A tile / column within B row
    const int lk = lane >> 4;   // K half-select (K vs K+2)
    const int m0 = blockIdx.x * 64 + waveId * 16;

    v8f acc[8];
    const v8f zero = {0.f, 0.f, 0.f, 0.f, 0.f, 0.f, 0.f, 0.f};
#pragma unroll
    for (int nt = 0; nt < 8; ++nt) acc[nt] = zero;

    const float* xrow = x + (size_t)(m0 + m) * FDIM;
    __builtin_prefetch(xrow, 0, 1);   // global_prefetch_b8

    for (int k0 = 0; k0 < FDIM; k0 += 4) {
        // A fragment 16x4 f32: lane(lk,m) holds x[m0+m][k0+2lk .. +1]
        // contiguous & 8B-aligned -> single global_load_b64
        const v2f afrag = *(const v2f*)(xrow + k0 + 2 * lk);
        const int kp = (k0 >> 1) + lk;
#pragma unroll
        for (int nt = 0; nt < 8; ++nt) {
            // B fragment 4x16 f32: one ds_load_b64, already paired
            const v2f bfrag = Wp[kp * FDIM + nt * 16 + m];
            acc[nt] = __builtin_amdgcn_wmma_f32_16x16x4_f32(
                false, afrag, false, bfrag, (short)0, acc[nt], false, false);
        }
    }

    // C/D layout: VGPR j -> row m0 + j + 8*lk, col n0 + m
#pragma unroll
    for (int nt = 0; nt < 8; ++nt) {
        const int n0 = nt * 16;
#pragma unroll
        for (int j = 0; j < 8; ++j)
            h[(size_t)(m0 + j + 8 * lk) * FDIM + n0 + m] = acc[nt][j];
    }
}

// ---------- 2) per-node scores: s1 = h@a_src, s2 = h@a_dst (one wave/node) ----------
__global__ void gat_scores(const float* __restrict__ h, const float* __restrict__ a,
                           float* __restrict__ s1, float* __restrict__ s2) {
    const int node = (blockIdx.x * blockDim.x + threadIdx.x) >> 5;
    const int lane = threadIdx.x & 31;
    if (node >= NODES) return;
    const float4 hv  = *(const float4*)(h + (size_t)node * FDIM + lane * 4);
    const float4 as  = *(const float4*)(a + lane * 4);
    const float4 ad  = *(const float4*)(a + FDIM + lane * 4);
    float d1 = hv.x * as.x + hv.y * as.y + hv.z * as.z + hv.w * as.w;
    float d2 = hv.x * ad.x + hv.y * ad.y + hv.z * ad.z + hv.w * ad.w;
#pragma unroll
    for (int off = 16; off > 0; off >>= 1) {   // wave32 reduction
        d1 += __shfl_xor(d1, off, 32);
        d2 += __shfl_xor(d2, off, 32);
    }
    if (lane == 0) { s1[node] = d1; s2[node] = d2; }
}

// ---------- 3) column means of h (for rows with no edges: uniform softmax) ----------
__global__ void gat_colsum(const float* __restrict__ h, float* __restrict__ colsum) {
    const int t  = threadIdx.x;              // 128 threads = one column each
    const int r0 = blockIdx.x * (NODES / gridDim.x);
    const int r1 = r0 + (NODES / gridDim.x);
    float acc = 0.0f;
    for (int r = r0; r < r1; ++r) acc += h[(size_t)r * FDIM + t];
    unsafeAtomicAdd(&colsum[t], acc);
}

// ---------- 4) edge pass 1: e = leaky_relu(s1[src]+s2[dst]); row max ----------
__global__ void gat_edge_max(const long long* __restrict__ ei,
                             const float* __restrict__ s1, const float* __restrict__ s2,
                             float* __restrict__ ework, unsigned* __restrict__ rowmax) {
    const int k = blockIdx.x * blockDim.x + threadIdx.x;
    if (k >= NEDGE) return;
    const int s = (int)ei[k];
    const int t = (int)ei[NEDGE + k];
    float e = s1[s] + s2[t];
    e = (e > 0.0f) ? e : ALPHA * e;
    ework[k] = e;
    atomicMax(&rowmax[s], fkey(e));
}

// ---------- 5) edge pass 2: w = exp(e - rowmax); row sum ----------
__global__ void gat_edge_exp(const long long* __restrict__ ei,
                             float* __restrict__ ework,
                             const unsigned* __restrict__ rowmax,
                             float* __restrict__ rowsum) {
    const int k = blockIdx.x * blockDim.x + threadIdx.x;
    if (k >= NEDGE) return;
    const int s = (int)ei[k];
    const float w = __expf(ework[k] - fdec(rowmax[s]));
    ework[k] = w;
    unsafeAtomicAdd(&rowsum[s], w);   // global_atomic_add_f32
}

// ---------- 6) aggregate: out[src] += (w/rowsum[src]) * h[dst]  (one wave/edge) ----------
__global__ void gat_aggregate(const long long* __restrict__ ei,
                              const float* __restrict__ h,
                              const float* __restrict__ ework,
                              const float* __restrict__ rowsum,
                              float* __restrict__ out) {
    const int k    = (blockIdx.x * blockDim.x + threadIdx.x) >> 5;
    const int lane = threadIdx.x & 31;
    if (k >= NEDGE) return;
    const int s = (int)ei[k];
    const int t = (int)ei[NEDGE + k];
    const float alpha = ework[k] / rowsum[s];              // scalar per wave
    const float4 hv = *(const float4*)(h + (size_t)t * FDIM + lane * 4);
    float* o = out + (size_t)s * FDIM + lane * 4;
    unsafeAtomicAdd(o + 0, alpha * hv.x);
    unsafeAtomicAdd(o + 1, alpha * hv.y);
    unsafeAtomicAdd(o + 2, alpha * hv.z);
    unsafeAtomicAdd(o + 3, alpha * hv.w);
}

// ---------- 7) finalize: empty rows -> colmean; add bias ----------
__global__ void gat_finalize(float* __restrict__ out,
                             const unsigned* __restrict__ rowmax,
                             const float* __restrict__ colsum,
                             const float* __restrict__ bias) {
    const int n = blockIdx.x;
    const int t = threadIdx.x;
    const size_t idx = (size_t)n * FDIM + t;
    float v = (rowmax[n] == 0u) ? colsum[t] * (1.0f / (float)NODES) : out[idx];
    out[idx] = v + bias[t];
}

extern "C" void kernel_launch(void* const* d_in, const int* in_sizes, int n_in,
                              void* d_out, int out_size, void* d_ws, size_t ws_size,
                              hipStream_t stream) {
    const float*     x    = (const float*)d_in[0];
    const long long* ei   = (const long long*)d_in[1];   // int64 edge_index (2, E)
    const float*     W    = (const float*)d_in[2];
    const float*     a    = (const float*)d_in[3];
    const float*     bias = (const float*)d_in[4];
    float* out = (float*)d_out;

    // workspace carve-out (~5.4 MB)
    char* ws = (char*)d_ws;
    float*    h      = (float*)ws;                              // 4 MB
    float*    s1     = (float*)(ws + ((size_t)NODES * FDIM * 4));
    float*    s2     = s1 + NODES;
    float*    ework  = s2 + NODES;                              // 1 MB
    float*    rowsum = ework + NEDGE;
    unsigned* rowmax = (unsigned*)(rowsum + NODES);
    float*    colsum = (float*)(rowmax + NODES);

    gat_zero<<<(NODES * FDIM + 255) / 256, 256, 0, stream>>>(out, rowmax, rowsum, colsum);
    gat_gemm<<<NODES / 64, 128, 0, stream>>>(x, W, h);
    gat_scores<<<(NODES * 32) / 256, 256, 0, stream>>>(h, a, s1, s2);
    gat_colsum<<<64, FDIM, 0, stream>>>(h, colsum);
    gat_edge_max<<<NEDGE / 256, 256, 0, stream>>>(ei, s1, s2, ework, rowmax);
    gat_edge_exp<<<NEDGE / 256, 256, 0, stream>>>(ei, ework, rowmax, rowsum);
    gat_aggregate<<<(NEDGE * 32) / 256, 256, 0, stream>>>(ei, h, ework, rowsum, out);
    gat_finalize<<<NODES, FDIM, 0, stream>>>(out, rowmax, colsum, bias);
}